// EpipolarLoss_26826365731209
// MI455X (gfx1250) — compile-verified
//
#include <hip/hip_runtime.h>
#include <math.h>

typedef float v2f __attribute__((ext_vector_type(2)));
typedef float v4f __attribute__((ext_vector_type(4)));
typedef float v8f __attribute__((ext_vector_type(8)));

namespace {
constexpr int Bn   = 32;
constexpr int H    = 384;
constexpr int Wd   = 512;
constexpr int N    = H * Wd;      // 196608 pixels per batch row
constexpr int HW   = N;           // channel stride
constexpr int CHW  = 3 * N;       // batch stride
constexpr int PID  = N / 2;       // 98304  (int(N*0.5))
constexpr int BPR  = 24;          // blocks per row (N % BPR == 0)
constexpr int CHUNK = N / BPR;    // 8192 pixels per block
constexpr int NBINS = 2048;
constexpr float CHI = (float)( 1.0 - 1e-7);  // match python float64->f32 constant
constexpr float CLO = (float)(-1.0 + 1e-7);
constexpr float THRESH = 0.1f;
}

// ---------------------------------------------------------------------------
// Deterministic wave32 sum using V_WMMA_F32_16X16X4_F32.
// A (16x4 f32, 2 VGPRs): lanes0-15 v0=K0, v1=K1 ; lanes16-31 v0=K2, v1=K3.
// We put x in v0 and 0 in v1 -> rowsum[m] = x[m] + x[m+16].
// B = all ones (layout irrelevant). D (16x16): vgpr r, lanes0-15 -> M=r,
// lanes16-31 -> M=r+8. Sum of the 8 D vgprs gives half-sum per lane group;
// shfl_xor(16) combines halves. Fixed order -> deterministic.
// ---------------------------------------------------------------------------
__device__ __forceinline__ float wave_sum_wmma(float x) {
  v2f a; a.x = x;    a.y = 0.0f;
  v2f b; b.x = 1.0f; b.y = 1.0f;
  v8f c = {};
  c = __builtin_amdgcn_wmma_f32_16x16x4_f32(false, a, false, b, (short)0, c,
                                            false, false);
  float s = c[0] + c[1] + c[2] + c[3] + c[4] + c[5] + c[6] + c[7];
  s += __shfl_xor(s, 16, 32);
  return s;  // every lane holds the full 32-lane sum
}

__device__ __forceinline__ float theta_of(float x0, float x1, float x2,
                                          float w0, float w1, float w2) {
  float m  = sqrtf(x0 * x0 + x1 * x1 + x2 * x2);
  float dp = (x0 / m) * w0 + (x1 / m) * w1 + (x2 / m) * w2;
  dp = fminf(fmaxf(dp, CLO), CHI);
  return acosf(dp);   // clamp guarantees the "safe" branch of acos_safe
}

// --------------------------- setup / clear kernels -------------------------
__global__ void init_state_k(unsigned* prefix, unsigned* tgt, unsigned* cnt) {
  int t = threadIdx.x;
  if (t < Bn) { prefix[t] = 0u; tgt[t] = (unsigned)PID; cnt[t] = 0u; }
}

__global__ void clear_hist_k(unsigned* hist) {
  int i = blockIdx.x * 256 + threadIdx.x;   // <<<256,256>>> covers Bn*NBINS
  if (i < Bn * NBINS) hist[i] = 0u;
}

// ---------------- pass 1: compute theta, store, histogram bits[31:21] ------
__global__ __launch_bounds__(256)
void theta_hist1_k(const float* __restrict__ yhat, const float* __restrict__ y,
                   float* __restrict__ theta, unsigned* __restrict__ hist) {
  __shared__ unsigned lh[NBINS];
  const int b = blockIdx.y;
  for (int i = threadIdx.x; i < NBINS; i += 256) lh[i] = 0u;
  __syncthreads();

  const float* yh = yhat + (size_t)b * CHW;
  const float* yy = y    + (size_t)b * CHW;
  float*       th = theta + (size_t)b * N;
  const int base = blockIdx.x * CHUNK;

  for (int i = threadIdx.x * 4; i < CHUNK; i += 256 * 4) {
    const int p = base + i;
    // inputs are read exactly once: non-temporal so L2 stays free for theta
    v4f a0 = __builtin_nontemporal_load((const v4f*)(yh + p));
    v4f a1 = __builtin_nontemporal_load((const v4f*)(yh + HW + p));
    v4f a2 = __builtin_nontemporal_load((const v4f*)(yh + 2 * HW + p));
    v4f b0 = __builtin_nontemporal_load((const v4f*)(yy + p));
    v4f b1 = __builtin_nontemporal_load((const v4f*)(yy + HW + p));
    v4f b2 = __builtin_nontemporal_load((const v4f*)(yy + 2 * HW + p));
    v4f t;
    t.x = theta_of(a0.x, a1.x, a2.x, b0.x, b1.x, b2.x);
    t.y = theta_of(a0.y, a1.y, a2.y, b0.y, b1.y, b2.y);
    t.z = theta_of(a0.z, a1.z, a2.z, b0.z, b1.z, b2.z);
    t.w = theta_of(a0.w, a1.w, a2.w, b0.w, b1.w, b2.w);
    *(v4f*)(th + p) = t;
    atomicAdd(&lh[__float_as_uint(t.x) >> 21], 1u);
    atomicAdd(&lh[__float_as_uint(t.y) >> 21], 1u);
    atomicAdd(&lh[__float_as_uint(t.z) >> 21], 1u);
    atomicAdd(&lh[__float_as_uint(t.w) >> 21], 1u);
  }
  __syncthreads();
  for (int i = threadIdx.x; i < NBINS; i += 256)
    if (lh[i]) atomicAdd(&hist[b * NBINS + i], lh[i]);
}

// ---------------- refinement histogram passes (bits [20:10], [9:0]) --------
__global__ __launch_bounds__(256)
void theta_histN_k(const float* __restrict__ theta,
                   const unsigned* __restrict__ prefix,
                   unsigned* __restrict__ hist,
                   int shift, int filtshift, unsigned binmask) {
  __shared__ unsigned lh[NBINS];
  const int b = blockIdx.y;
  const unsigned pref = prefix[b] >> filtshift;
  for (int i = threadIdx.x; i < NBINS; i += 256) lh[i] = 0u;
  __syncthreads();

  const float* th = theta + (size_t)b * N;
  const int base = blockIdx.x * CHUNK;
  for (int i = threadIdx.x * 4; i < CHUNK; i += 256 * 4) {
    v4f t = *(const v4f*)(th + base + i);
    unsigned k;
    k = __float_as_uint(t.x);
    if ((k >> filtshift) == pref) atomicAdd(&lh[(k >> shift) & binmask], 1u);
    k = __float_as_uint(t.y);
    if ((k >> filtshift) == pref) atomicAdd(&lh[(k >> shift) & binmask], 1u);
    k = __float_as_uint(t.z);
    if ((k >> filtshift) == pref) atomicAdd(&lh[(k >> shift) & binmask], 1u);
    k = __float_as_uint(t.w);
    if ((k >> filtshift) == pref) atomicAdd(&lh[(k >> shift) & binmask], 1u);
  }
  __syncthreads();
  for (int i = threadIdx.x; i < NBINS; i += 256)
    if (lh[i]) atomicAdd(&hist[b * NBINS + i], lh[i]);
}

// ---------------- per-row histogram scan: pick bin holding target ----------
__global__ void scan_hist_k(const unsigned* __restrict__ hist,
                            unsigned* prefix, unsigned* tgt, unsigned* cnt,
                            int shift, int nbins) {
  const int b = threadIdx.x;
  if (b >= Bn) return;
  const unsigned* h = hist + b * NBINS;
  unsigned t = tgt[b];
  unsigned cumB = 0;
  int sel = 0;
  for (int i = 0; i < nbins; ++i) {
    unsigned c = h[i];
    if (t < cumB + c) { sel = i; break; }
    cumB += c;
  }
  prefix[b] |= (unsigned)sel << shift;
  tgt[b] = t - cumB;
  cnt[b] += cumB;
}

// ---------------- conditional sums over theta ------------------------------
__global__ __launch_bounds__(256)
void sums_k(const float* __restrict__ theta, const unsigned* __restrict__ vbits,
            float* __restrict__ pltv, float* __restrict__ pltt) {
  const int b = blockIdx.y;
  const unsigned vb = vbits[b];          // bits of the PID-th order statistic
  const float* th = theta + (size_t)b * N;
  const int base = blockIdx.x * CHUNK;
  float s1 = 0.0f, s2 = 0.0f;
  for (int i = threadIdx.x * 4; i < CHUNK; i += 256 * 4) {
    v4f t = *(const v4f*)(th + base + i);
    s1 += (__float_as_uint(t.x) < vb) ? t.x : 0.0f;
    s1 += (__float_as_uint(t.y) < vb) ? t.y : 0.0f;
    s1 += (__float_as_uint(t.z) < vb) ? t.z : 0.0f;
    s1 += (__float_as_uint(t.w) < vb) ? t.w : 0.0f;
    s2 += (t.x < THRESH) ? t.x : 0.0f;
    s2 += (t.y < THRESH) ? t.y : 0.0f;
    s2 += (t.z < THRESH) ? t.z : 0.0f;
    s2 += (t.w < THRESH) ? t.w : 0.0f;
  }
  // uniform control flow up to here -> EXEC all ones for WMMA
  float r1 = wave_sum_wmma(s1);
  float r2 = wave_sum_wmma(s2);
  __shared__ float w1[8], w2[8];
  const int wid = threadIdx.x >> 5, lane = threadIdx.x & 31;
  if (lane == 0) { w1[wid] = r1; w2[wid] = r2; }
  __syncthreads();
  if (threadIdx.x == 0) {
    float a1 = 0.0f, a2 = 0.0f;
    for (int i = 0; i < 8; ++i) { a1 += w1[i]; a2 += w2[i]; }
    pltv[b * BPR + blockIdx.x] = a1;
    pltt[b * BPR + blockIdx.x] = a2;
  }
}

// ---------------- final combine: 32 rows -> scalar -------------------------
__global__ void final_k(const float* __restrict__ pltv,
                        const float* __restrict__ pltt,
                        const unsigned* __restrict__ vbits,
                        const unsigned* __restrict__ cnt,
                        float* __restrict__ out) {
  const int b = threadIdx.x;   // exactly one wave of 32
  float s1 = 0.0f, s2 = 0.0f;
  for (int i = 0; i < BPR; ++i) {          // fixed order -> deterministic
    s1 += pltv[b * BPR + i];
    s2 += pltt[b * BPR + i];
  }
  const float v = __uint_as_float(vbits[b]);
  const float ties = v * (float)(PID - (int)cnt[b]);  // exact tie handling
  const float contrib = (v < THRESH) ? (s1 + ties) : s2;
  const float total = wave_sum_wmma(contrib);
  if (b == 0) out[0] = total;
}

// ---------------------------------------------------------------------------
extern "C" void kernel_launch(void* const* d_in, const int* in_sizes, int n_in,
                              void* d_out, int out_size, void* d_ws,
                              size_t ws_size, hipStream_t stream) {
  const float* yhat = (const float*)d_in[0];
  const float* y    = (const float*)d_in[1];
  float* out = (float*)d_out;

  // workspace layout (~25.5 MB total)
  char* ws = (char*)d_ws;
  float*    theta  = (float*)ws;                               // Bn*N floats
  unsigned* hist   = (unsigned*)(ws + (size_t)Bn * N * 4);     // Bn*NBINS
  unsigned* prefix = hist + Bn * NBINS;                        // Bn
  unsigned* tgt    = prefix + Bn;                              // Bn
  unsigned* cnt    = tgt + Bn;                                 // Bn
  float*    pltv   = (float*)(cnt + Bn);                       // Bn*BPR
  float*    pltt   = pltv + Bn * BPR;                          // Bn*BPR

  const dim3 grid(BPR, Bn);

  init_state_k<<<1, 32, 0, stream>>>(prefix, tgt, cnt);

  // radix-select pass 1: bits [31:21]
  clear_hist_k<<<256, 256, 0, stream>>>(hist);
  theta_hist1_k<<<grid, 256, 0, stream>>>(yhat, y, theta, hist);
  scan_hist_k<<<1, 32, 0, stream>>>(hist, prefix, tgt, cnt, 21, 2048);

  // pass 2: bits [20:10]
  clear_hist_k<<<256, 256, 0, stream>>>(hist);
  theta_histN_k<<<grid, 256, 0, stream>>>(theta, prefix, hist, 10, 21, 2047u);
  scan_hist_k<<<1, 32, 0, stream>>>(hist, prefix, tgt, cnt, 10, 2048);

  // pass 3: bits [9:0]  -> prefix == exact bits of the PID-th value
  clear_hist_k<<<256, 256, 0, stream>>>(hist);
  theta_histN_k<<<grid, 256, 0, stream>>>(theta, prefix, hist, 0, 10, 1023u);
  scan_hist_k<<<1, 32, 0, stream>>>(hist, prefix, tgt, cnt, 0, 1024);

  // conditional sums + final combine
  sums_k<<<grid, 256, 0, stream>>>(theta, prefix, pltv, pltt);
  final_k<<<1, 32, 0, stream>>>(pltv, pltt, prefix, cnt, out);
}